// Decoder_5282809774314
// MI455X (gfx1250) — compile-verified
//
#include <hip/hip_runtime.h>
#include <stdint.h>

// Pointer-generator decoder step for MI455X (gfx1250, wave32).
//
// Roofline: the attention feature GEMM (65536x1024 @ 1024x1024, 137 GFLOP)
// is memory-bound on streaming h (268 MB fp32, ~11.5us at 23.3 TB/s HBM);
// f32 WMMA (V_WMMA_F32_16X16X4_F32) keeps full precision while making the
// matrix math free relative to bandwidth. tanh+dot(attn_v) is fused in the
// epilogue so the 268 MB 'feat' tensor never touches memory. The GEMM
// stages tiles into LDS with CDNA5 GLOBAL_LOAD_ASYNC_TO_LDS_B32/B64
// (ASYNCcnt-tracked, double-buffered: chunk k+1 copies overlap chunk k
// WMMAs, synced with s_wait_asynccnt + workgroup barrier). The context
// reduction re-reads h once more (another 268 MB pass). Everything else is
// tiny. No atomics anywhere -> bit-deterministic across replays.

#define Bq 8
#define Lq 8192
#define Hq 512
#define Eq 256
#define VOCAB 8192
#define TWO_H 1024
#define FOUR_H 2048
#define AW_STRIDE 2049   // attn_W row stride: (2H, 4H+1)

typedef float v2f __attribute__((ext_vector_type(2)));
typedef float v8f __attribute__((ext_vector_type(8)));

__device__ __forceinline__ float sigmoidf_(float x) { return 1.0f / (1.0f + __expf(-x)); }

// Low 32 bits of a flat shared pointer == hardware LDS offset (ISA aperture rule).
__device__ __forceinline__ unsigned lds_off_(const void* p) {
  return (unsigned)(uintptr_t)p;
}

// D = A(16x4) * B(4x16) + C  (f32 WMMA, wave32)
#define WMMA_F32(a, bb, cc) \
  __builtin_amdgcn_wmma_f32_16x16x4_f32(false, (a), false, (bb), (short)0, (cc), false, false)

// ---------------------------------------------------------------------------
// K1: x = [emb[y_prev], c_t_pre] @ feed_W.T + feed_b          (B,256)
// ---------------------------------------------------------------------------
__global__ void feed_kernel(const float* __restrict__ emb, const int* __restrict__ y_prev,
                            const float* __restrict__ c_t_pre, const float* __restrict__ feed_W,
                            const float* __restrict__ feed_b, float* __restrict__ x) {
  const int b = blockIdx.x, t = threadIdx.x;          // 256 threads
  __shared__ float cat[1280];
  const float* e = emb + (size_t)y_prev[b] * Eq;
  for (int j = t; j < 1280; j += 256)
    cat[j] = (j < Eq) ? e[j] : c_t_pre[b * TWO_H + (j - Eq)];
  __syncthreads();
  float s = feed_b[t];
  const float* w = feed_W + (size_t)t * 1280;
  for (int j = 0; j < 1280; ++j) s += cat[j] * w[j];
  x[b * Eq + t] = s;
}

// ---------------------------------------------------------------------------
// K2: gates = x@Wih.T + bih + s_h@Whh.T + bhh                  (B,2048)
// ---------------------------------------------------------------------------
__global__ void gates_kernel(const float* __restrict__ x, const float* __restrict__ s_h,
                             const float* __restrict__ Wih, const float* __restrict__ Whh,
                             const float* __restrict__ bih, const float* __restrict__ bhh,
                             float* __restrict__ gates) {
  const int b = blockIdx.x;
  const int g = blockIdx.y * 256 + threadIdx.x;
  __shared__ float xs[Eq];
  __shared__ float hs[Hq];
  for (int j = threadIdx.x; j < Eq; j += 256) xs[j] = x[b * Eq + j];
  for (int j = threadIdx.x; j < Hq; j += 256) hs[j] = s_h[b * Hq + j];
  __syncthreads();
  float s = bih[g] + bhh[g];
  const float* wi = Wih + (size_t)g * Eq;
  #pragma unroll 4
  for (int j = 0; j < Eq; ++j) s += xs[j] * wi[j];
  const float* wh = Whh + (size_t)g * Hq;
  #pragma unroll 4
  for (int j = 0; j < Hq; ++j) s += hs[j] * wh[j];
  gates[b * FOUR_H + g] = s;
}

// ---------------------------------------------------------------------------
// K3: LSTM cell elementwise -> h_new, c_new                    (B,512) each
// ---------------------------------------------------------------------------
__global__ void lstm_kernel(const float* __restrict__ gates, const float* __restrict__ s_c,
                            float* __restrict__ hn, float* __restrict__ cn) {
  const int b = blockIdx.x, k = threadIdx.x;          // 512 threads
  const float* gb = gates + b * FOUR_H;
  float ig = sigmoidf_(gb[k]);
  float fg = sigmoidf_(gb[Hq + k]);
  float gg = tanhf(gb[2 * Hq + k]);
  float og = sigmoidf_(gb[3 * Hq + k]);
  float c = fg * s_c[b * Hq + k] + ig * gg;
  cn[b * Hq + k] = c;
  hn[b * Hq + k] = og * tanhf(c);
}

// ---------------------------------------------------------------------------
// K4: sbias[b][m] = s_hat @ Ws.T + attn_b   (Ws = attn_W[:,1024:2048])
// ---------------------------------------------------------------------------
__global__ __launch_bounds__(1024) void sbias_kernel(const float* __restrict__ hn,
                                                     const float* __restrict__ cn,
                                                     const float* __restrict__ attn_W,
                                                     const float* __restrict__ attn_b,
                                                     float* __restrict__ sbias) {
  const int b = blockIdx.x, m = threadIdx.x;          // 1024 threads
  __shared__ float sh[TWO_H];
  sh[m] = (m < Hq) ? hn[b * Hq + m] : cn[b * Hq + (m - Hq)];
  __syncthreads();
  float s = attn_b[m];
  const float* w = attn_W + (size_t)m * AW_STRIDE + TWO_H;
  #pragma unroll 4
  for (int j = 0; j < TWO_H; ++j) s += sh[j] * w[j];
  sbias[b * TWO_H + m] = s;
}

// ---------------------------------------------------------------------------
// K5: fused attention scores via f32 WMMA + async-to-LDS double buffering.
//   scores_part[stripe][b][l] = sum_{m in stripe} tanh( (h@Wh.T)[l,m]
//                                   + sbias[b][m] + cov[b][l]*wc[m] ) * v[m]
// Grid: (256 row-tiles of 256 l-rows, 4 m-stripes of 256), 1024 threads.
// Per WG: 32 waves in an 8x4 grid; wave = 32 rows x 64 m = 8 C tiles.
// LDS: 2x(A 256x16 + B 256x16), padded stride 18 dwords, 72 KB total.
// Copies use GLOBAL_LOAD_ASYNC_TO_LDS_{B64,B32}; 6 async instrs per wave per
// chunk, in-order completion -> s_wait_asynccnt 6 after issuing the next
// chunk guarantees the current chunk is resident (then barrier).
// ---------------------------------------------------------------------------
#define APAD 18
#define CH_F (256 * APAD)   // floats per A or B chunk buffer
__global__ __launch_bounds__(1024) void score_kernel(
    const float* __restrict__ h, const float* __restrict__ attn_W,
    const float* __restrict__ sbias, const float* __restrict__ attn_v,
    const float* __restrict__ coverage, float* __restrict__ scores_part) {
  __shared__ float lds[4 * CH_F];                     // A0,B0,A1,B1 = 72 KB

  const int t = threadIdx.x;
  const int rtg = blockIdx.x;                         // 0..255
  const int stripe = blockIdx.y;                      // 0..3
  const int b = rtg >> 5;
  const int l0 = (rtg & 31) * 256;
  const int m0 = stripe * 256;

  const int w = t >> 5;
  const int lane = t & 31;
  const int wr = w >> 2;                              // 0..7 (row waves)
  const int wc = w & 3;                               // 0..3 (col waves)
  const int ml = lane & 15;
  const int kh = lane >> 4;

  const v8f zacc = {0.f, 0.f, 0.f, 0.f, 0.f, 0.f, 0.f, 0.f};
  v8f c[2][4];
  #pragma unroll
  for (int rt = 0; rt < 2; ++rt)
    #pragma unroll
    for (int ct = 0; ct < 4; ++ct) c[rt][ct] = zacc;

  const float* __restrict__ hrow = h + ((size_t)b * Lq + l0) * (size_t)TWO_H;

  // Issue one chunk's async copies (per wave: 2x B64 for A, 4x B32 for B).
  auto issue_chunk = [&](int kc, float* Al, float* Bl) {
    const int k0 = kc * 16;
    #pragma unroll
    for (int i = 0; i < 2; ++i) {                     // A: 256 rows x 16 k, b64/lane
      int q = t + i * 1024;
      int r = q >> 3, kk = (q & 7) * 2;
      const float* g = hrow + (size_t)r * TWO_H + k0 + kk;
      unsigned l = lds_off_(Al + r * APAD + kk);
      asm volatile("global_load_async_to_lds_b64 %0, %1, off"
                   :: "v"(l), "v"(g) : "memory");
    }
    #pragma unroll
    for (int i = 0; i < 4; ++i) {                     // B: Wh rows m0.. (stride 2049), b32/lane
      int q = t + i * 1024;
      int r = q >> 4, kk = q & 15;
      const float* g = attn_W + (size_t)(m0 + r) * AW_STRIDE + k0 + kk;
      unsigned l = lds_off_(Bl + r * APAD + kk);
      asm volatile("global_load_async_to_lds_b32 %0, %1, off"
                   :: "v"(l), "v"(g) : "memory");
    }
  };

  issue_chunk(0, lds, lds + CH_F);                    // prologue: chunk 0 -> buf0

  for (int kc = 0; kc < 64; ++kc) {                   // K = 1024, chunks of 16
    const int cur = kc & 1;
    float* Ac = lds + cur * 2 * CH_F;
    float* Bc = Ac + CH_F;
    if (kc < 63) {
      float* An = lds + (1 - cur) * 2 * CH_F;         // last read fenced by prev barrier
      issue_chunk(kc + 1, An, An + CH_F);
      asm volatile("s_wait_asynccnt 0x6" ::: "memory");   // chunk kc resident
    } else {
      asm volatile("s_wait_asynccnt 0x0" ::: "memory");
    }
    __syncthreads();                                  // all waves' data visible

    // A frag lane view: M = ml, K pair = {2*kh, 2*kh+1}  (ISA 16x4 f32 layout)
    const float* Afrag0 = Ac + (wr * 32 + ml) * APAD;
    const float* Afrag1 = Afrag0 + 16 * APAD;
    // B frag lane view: N = ml, K pair striped like A   (B = Wh^T tile)
    const float* Bfrag = Bc + (wc * 64 + ml) * APAD;

    #pragma unroll
    for (int ks = 0; ks < 4; ++ks) {
      const int ko = ks * 4 + kh * 2;
      v2f a0 = *(const v2f*)(Afrag0 + ko);
      v2f a1 = *(const v2f*)(Afrag1 + ko);
      v2f bf[4];
      #pragma unroll
      for (int ct = 0; ct < 4; ++ct) bf[ct] = *(const v2f*)(Bfrag + ct * 16 * APAD + ko);
      #pragma unroll
      for (int ct = 0; ct < 4; ++ct) {
        c[0][ct] = WMMA_F32(a0, bf[ct], c[0][ct]);
        c[1][ct] = WMMA_F32(a1, bf[ct], c[1][ct]);
      }
    }
    __syncthreads();                                  // fence before buffer reuse
  }

  // Epilogue: feat = tanh(C + sbias[m] + cov[l]*wc[m]); score += feat*v[m].
  float sb[4], wcv[4], vv[4];
  #pragma unroll
  for (int ct = 0; ct < 4; ++ct) {
    int m = m0 + wc * 64 + ct * 16 + ml;
    sb[ct] = sbias[b * TWO_H + m];
    wcv[ct] = attn_W[(size_t)m * AW_STRIDE + FOUR_H];  // wc column (4H)
    vv[ct] = attn_v[m];
  }
  const float* covp = coverage + (size_t)b * Lq + l0;
  float* red = lds;                                    // [256][4], reuse LDS
  #pragma unroll
  for (int rt = 0; rt < 2; ++rt)
    #pragma unroll
    for (int j = 0; j < 8; ++j) {
      // C/D layout: VGPR j -> M = j + 8*kh, N = ml
      int row = wr * 32 + rt * 16 + kh * 8 + j;
      float cv = covp[row];
      float s = 0.f;
      #pragma unroll
      for (int ct = 0; ct < 4; ++ct)
        s += tanhf(c[rt][ct][j] + sb[ct] + cv * wcv[ct]) * vv[ct];
      // reduce over 16 lanes of this half (wave32 butterflies stay in-half)
      #pragma unroll
      for (int off = 1; off < 16; off <<= 1) s += __shfl_xor(s, off, 32);
      if (ml == 0) red[row * 4 + wc] = s;
    }
  __syncthreads();
  if (t < 256) {
    float s = red[t * 4] + red[t * 4 + 1] + red[t * 4 + 2] + red[t * 4 + 3];
    scores_part[((size_t)stripe * Bq + b) * Lq + l0 + t] = s;
  }
}

// ---------------------------------------------------------------------------
// K6: attn = softmax(sum of stripe partials); coverage_next = coverage + attn
// ---------------------------------------------------------------------------
__global__ __launch_bounds__(1024) void attn_softmax_kernel(
    const float* __restrict__ scores_part, const float* __restrict__ coverage,
    float* __restrict__ attn_out, float* __restrict__ cov_out) {
  const int b = blockIdx.x, t = threadIdx.x;          // 1024 threads, 8 l each
  __shared__ float red[1024];
  float s[8];
  float mx = -1e30f;
  #pragma unroll
  for (int i = 0; i < 8; ++i) {
    int l = t + i * 1024;
    float v = 0.f;
    #pragma unroll
    for (int st = 0; st < 4; ++st) v += scores_part[((size_t)st * Bq + b) * Lq + l];
    s[i] = v;
    mx = fmaxf(mx, v);
  }
  red[t] = mx; __syncthreads();
  for (int o = 512; o > 0; o >>= 1) { if (t < o) red[t] = fmaxf(red[t], red[t + o]); __syncthreads(); }
  mx = red[0]; __syncthreads();
  float sum = 0.f;
  #pragma unroll
  for (int i = 0; i < 8; ++i) { s[i] = __expf(s[i] - mx); sum += s[i]; }
  red[t] = sum; __syncthreads();
  for (int o = 512; o > 0; o >>= 1) { if (t < o) red[t] += red[t + o]; __syncthreads(); }
  const float inv = 1.f / red[0];
  #pragma unroll
  for (int i = 0; i < 8; ++i) {
    int l = t + i * 1024;
    float a = s[i] * inv;
    attn_out[(size_t)b * Lq + l] = a;
    cov_out[(size_t)b * Lq + l] = coverage[(size_t)b * Lq + l] + a;
  }
}

// ---------------------------------------------------------------------------
// K7/K8: c_t = attn @ h  (two-stage deterministic reduction over L)
// ---------------------------------------------------------------------------
__global__ void ctx_part_kernel(const float* __restrict__ h, const float* __restrict__ attn,
                                float* __restrict__ cp) {
  const int ch = blockIdx.x, b = blockIdx.y, t = threadIdx.x;  // 64 chunks x 256 thr
  const int n = t * 4;
  const float* hp = h + ((size_t)b * Lq + (size_t)ch * 128) * TWO_H + n;
  const float* ap = attn + (size_t)b * Lq + ch * 128;
  float4 acc = make_float4(0.f, 0.f, 0.f, 0.f);
  for (int l = 0; l < 128; ++l) {
    float a = ap[l];
    float4 hv = *(const float4*)(hp + (size_t)l * TWO_H);
    acc.x += a * hv.x; acc.y += a * hv.y; acc.z += a * hv.z; acc.w += a * hv.w;
  }
  *(float4*)(cp + ((size_t)ch * Bq + b) * TWO_H + n) = acc;
}

__global__ void ctx_reduce_kernel(const float* __restrict__ cp, float* __restrict__ ct) {
  const int b = blockIdx.x, t = threadIdx.x;          // 256 threads
  const int n = t * 4;
  float4 acc = make_float4(0.f, 0.f, 0.f, 0.f);
  for (int ch = 0; ch < 64; ++ch) {
    float4 v = *(const float4*)(cp + ((size_t)ch * Bq + b) * TWO_H + n);
    acc.x += v.x; acc.y += v.y; acc.z += v.z; acc.w += v.w;
  }
  *(float4*)(ct + b * TWO_H + n) = acc;
}

// ---------------------------------------------------------------------------
// K9: p_gen = sigmoid([c_t, h_new, c_new, x] . pgen_W + pgen_b)
// ---------------------------------------------------------------------------
__global__ void pgen_kernel(const float* __restrict__ ct, const float* __restrict__ hn,
                            const float* __restrict__ cn, const float* __restrict__ x,
                            const float* __restrict__ pgen_W, const float* __restrict__ pgen_b,
                            float* __restrict__ pg) {
  const int b = blockIdx.x, t = threadIdx.x;          // 256 threads
  __shared__ float red[256];
  float s = 0.f;
  for (int j = t; j < 2304; j += 256) {
    float v;
    if (j < 1024)      v = ct[b * TWO_H + j];
    else if (j < 1536) v = hn[b * Hq + (j - 1024)];
    else if (j < 2048) v = cn[b * Hq + (j - 1536)];
    else               v = x[b * Eq + (j - 2048)];
    s += v * pgen_W[j];
  }
  red[t] = s; __syncthreads();
  for (int o = 128; o > 0; o >>= 1) { if (t < o) red[t] += red[t + o]; __syncthreads(); }
  if (t == 0) pg[b] = sigmoidf_(red[0] + pgen_b[0]);
}

// ---------------------------------------------------------------------------
// K10: t = [h_new, c_t] @ V_W.T + V_b                          (B,512)
// ---------------------------------------------------------------------------
__global__ void tproj_kernel(const float* __restrict__ hn, const float* __restrict__ ct,
                             const float* __restrict__ V_W, const float* __restrict__ V_b,
                             float* __restrict__ tb) {
  const int b = blockIdx.x, t = threadIdx.x;          // 512 threads
  __shared__ float cat[1536];
  for (int j = t; j < 1536; j += 512) cat[j] = (j < Hq) ? hn[b * Hq + j] : ct[b * TWO_H + (j - Hq)];
  __syncthreads();
  float s = V_b[t];
  const float* w = V_W + (size_t)t * 1536;
  #pragma unroll 4
  for (int j = 0; j < 1536; ++j) s += cat[j] * w[j];
  tb[b * Hq + t] = s;
}

// ---------------------------------------------------------------------------
// K11: logits = t @ Vp_W.T + Vp_b                              (B,8192)
// ---------------------------------------------------------------------------
__global__ void logits_kernel(const float* __restrict__ tb, const float* __restrict__ Vp_W,
                              const float* __restrict__ Vp_b, float* __restrict__ lg) {
  const int b = blockIdx.x;
  const int v = blockIdx.y * 512 + threadIdx.x;       // 16 x 512
  __shared__ float ts[Hq];
  ts[threadIdx.x] = tb[b * Hq + threadIdx.x];
  __syncthreads();
  float s = Vp_b[v];
  const float* w = Vp_W + (size_t)v * Hq;
  #pragma unroll 4
  for (int j = 0; j < Hq; ++j) s += ts[j] * w[j];
  lg[(size_t)b * VOCAB + v] = s;
}

// ---------------------------------------------------------------------------
// K12: P_vocab = softmax(logits); P_w = pg*P_vocab + (1-pg)*attn
// ---------------------------------------------------------------------------
__global__ __launch_bounds__(1024) void vocab_mix_kernel(
    const float* __restrict__ lg, const float* __restrict__ attn,
    const float* __restrict__ pg, float* __restrict__ pw) {
  const int b = blockIdx.x, t = threadIdx.x;
  __shared__ float red[1024];
  float s[8];
  float mx = -1e30f;
  #pragma unroll
  for (int i = 0; i < 8; ++i) { s[i] = lg[(size_t)b * VOCAB + t + i * 1024]; mx = fmaxf(mx, s[i]); }
  red[t] = mx; __syncthreads();
  for (int o = 512; o > 0; o >>= 1) { if (t < o) red[t] = fmaxf(red[t], red[t + o]); __syncthreads(); }
  mx = red[0]; __syncthreads();
  float sum = 0.f;
  #pragma unroll
  for (int i = 0; i < 8; ++i) { s[i] = __expf(s[i] - mx); sum += s[i]; }
  red[t] = sum; __syncthreads();
  for (int o = 512; o > 0; o >>= 1) { if (t < o) red[t] += red[t + o]; __syncthreads(); }
  const float inv = 1.f / red[0];
  const float p = pg[b];
  #pragma unroll
  for (int i = 0; i < 8; ++i) {
    int v = t + i * 1024;
    pw[(size_t)b * VOCAB + v] = p * (s[i] * inv) + (1.f - p) * attn[(size_t)b * Lq + v];
  }
}

// ---------------------------------------------------------------------------
extern "C" void kernel_launch(void* const* d_in, const int* in_sizes, int n_in,
                              void* d_out, int out_size, void* d_ws, size_t ws_size,
                              hipStream_t stream) {
  (void)in_sizes; (void)n_in; (void)out_size; (void)ws_size;
  const float* h        = (const float*)d_in[0];
  const float* s_h      = (const float*)d_in[1];
  const float* s_c      = (const float*)d_in[2];
  const int*   y_prev   = (const int*)  d_in[3];
  const float* c_t_pre  = (const float*)d_in[4];
  const float* coverage = (const float*)d_in[5];
  const float* emb      = (const float*)d_in[6];
  const float* feed_W   = (const float*)d_in[7];
  const float* feed_b   = (const float*)d_in[8];
  const float* Wih      = (const float*)d_in[9];
  const float* Whh      = (const float*)d_in[10];
  const float* bih      = (const float*)d_in[11];
  const float* bhh      = (const float*)d_in[12];
  const float* attn_W   = (const float*)d_in[13];
  const float* attn_b   = (const float*)d_in[14];
  const float* attn_v   = (const float*)d_in[15];
  const float* pgen_W   = (const float*)d_in[16];
  const float* pgen_b   = (const float*)d_in[17];
  const float* V_W      = (const float*)d_in[18];
  const float* V_b      = (const float*)d_in[19];
  const float* Vp_W     = (const float*)d_in[20];
  const float* Vp_b     = (const float*)d_in[21];

  // Output layout (flat, in return order)
  float* out = (float*)d_out;
  float* PW   = out + 0;        // (8,8192)
  float* ATTN = out + 65536;    // (8,8192)
  float* CT   = out + 131072;   // (8,1024)
  float* COV  = out + 139264;   // (8,8192)
  float* HN   = out + 204800;   // (8,512)
  float* CN   = out + 208896;   // (8,512)
  float* PG   = out + 212992;   // (8,1)

  // Workspace layout (floats); total ~3.4 MB
  float* ws      = (float*)d_ws;
  float* ws_x    = ws + 0;        // 2048
  float* ws_g    = ws + 2048;     // 16384
  float* ws_sb   = ws + 18432;    // 8192
  float* ws_sp   = ws + 26624;    // 262144 : 4 stripes x (8,8192) score partials
  float* ws_cp   = ws + 288768;   // 524288 : 64 chunks x (8,1024) ctx partials
  float* ws_t    = ws + 813056;   // 4096
  float* ws_lg   = ws + 817152;   // 65536

  feed_kernel        <<<dim3(Bq),      dim3(Eq),   0, stream>>>(emb, y_prev, c_t_pre, feed_W, feed_b, ws_x);
  gates_kernel       <<<dim3(Bq, 8),   dim3(256),  0, stream>>>(ws_x, s_h, Wih, Whh, bih, bhh, ws_g);
  lstm_kernel        <<<dim3(Bq),      dim3(Hq),   0, stream>>>(ws_g, s_c, HN, CN);
  sbias_kernel       <<<dim3(Bq),      dim3(1024), 0, stream>>>(HN, CN, attn_W, attn_b, ws_sb);
  score_kernel       <<<dim3(256, 4),  dim3(1024), 0, stream>>>(h, attn_W, ws_sb, attn_v, coverage, ws_sp);
  attn_softmax_kernel<<<dim3(Bq),      dim3(1024), 0, stream>>>(ws_sp, coverage, ATTN, COV);
  ctx_part_kernel    <<<dim3(64, Bq),  dim3(256),  0, stream>>>(h, ATTN, ws_cp);
  ctx_reduce_kernel  <<<dim3(Bq),      dim3(256),  0, stream>>>(ws_cp, CT);
  pgen_kernel        <<<dim3(Bq),      dim3(256),  0, stream>>>(CT, HN, CN, ws_x, pgen_W, pgen_b, PG);
  tproj_kernel       <<<dim3(Bq),      dim3(512),  0, stream>>>(HN, CT, V_W, V_b, ws_t);
  logits_kernel      <<<dim3(Bq, 16),  dim3(512),  0, stream>>>(ws_t, Vp_W, Vp_b, ws_lg);
  vocab_mix_kernel   <<<dim3(Bq),      dim3(1024), 0, stream>>>(ws_lg, ATTN, PG, PW);
}